// AlignmentForce_43241730736139
// MI455X (gfx1250) — compile-verified
//
#include <hip/hip_runtime.h>
#include <hip/hip_bf16.h>

#define BLK 256
#define MAXC 16          // max chains supported (reference uses 4)
#define NQ (MAXC * 7)    // per-block partial slots for pocket-COM pass

// ---- optional CDNA5 async global->LDS staging (guarded; falls back cleanly)
#ifndef TRY_ASYNC_LDS
#define TRY_ASYNC_LDS 1
#endif
#if TRY_ASYNC_LDS && defined(__gfx1250__) && \
    __has_builtin(__builtin_amdgcn_global_load_async_to_lds_b32) && \
    __has_builtin(__builtin_amdgcn_s_wait_asynccnt)
#define HAVE_ASYNC_LDS 1
#else
#define HAVE_ASYNC_LDS 0
#endif

// hdr layout (floats, in d_ws):
//  [0..2]  origin   [3..5] F_mean   [6..8] torque_mean
//  [12..12+3*MAXC)  best_t[c][3]

__device__ __forceinline__ float wave_sum(float v) {
#pragma unroll
  for (int off = 16; off > 0; off >>= 1) v += __shfl_down(v, off, 32);
  return v;
}

// valid on thread 0; sm must hold >= 8 floats
__device__ __forceinline__ float block_sum(float v, float* sm) {
  const int lane = threadIdx.x & 31;
  const int wid  = threadIdx.x >> 5;
  v = wave_sum(v);
  if (lane == 0) sm[wid] = v;
  __syncthreads();
  if (wid == 0) {
    float x = (lane < (int)(blockDim.x >> 5)) ? sm[lane] : 0.0f;
    v = wave_sum(x);
  }
  __syncthreads();
  return v;
}

// Vectorized zero: one b128 store per lane (d_out is allocation-aligned).
__global__ void k_zero4(float4* __restrict__ out4, int n4) {
  const int i = blockIdx.x * BLK + threadIdx.x;
  if (i < n4) out4[i] = make_float4(0.f, 0.f, 0.f, 0.f);
}

// Scalar tail for the last (out_size % 4) elements.
__global__ void k_zero_tail(float* __restrict__ out, int start, int n) {
  const int i = start + (int)threadIdx.x;
  if (i < n) out[i] = 0.0f;
}

// Pass 1: per-chain sums of positions[poc_indices], ref_poc, and counts.
__global__ void k_pocket_com(const float* __restrict__ pos,
                             const float* __restrict__ refp,
                             const int*   __restrict__ pidx,
                             const int*   __restrict__ pcid,
                             float* __restrict__ partA, int P) {
  __shared__ float s[NQ];
#if HAVE_ASYNC_LDS
  __shared__ float s_ref[BLK * 3];
#endif
  const int t = threadIdx.x;
  const int p = blockIdx.x * BLK + t;
  const bool act = (p < P);

#if HAVE_ASYNC_LDS
  if (act) {  // stage coalesced ref_poc stream into LDS while we do the gather
    const float* g = refp + 3 * (size_t)p;
    __builtin_amdgcn_global_load_async_to_lds_b32((int*)(g + 0), (int*)&s_ref[3 * t + 0], 0, 0);
    __builtin_amdgcn_global_load_async_to_lds_b32((int*)(g + 1), (int*)&s_ref[3 * t + 1], 0, 0);
    __builtin_amdgcn_global_load_async_to_lds_b32((int*)(g + 2), (int*)&s_ref[3 * t + 2], 0, 0);
  }
#endif

  for (int i = t; i < NQ; i += BLK) s[i] = 0.0f;
  __syncthreads();

  int c = 0;
  float gx = 0.f, gy = 0.f, gz = 0.f, rx = 0.f, ry = 0.f, rz = 0.f;
  if (act) {
    c = pcid[p];
    const int gi = pidx[p];               // bug-faithful: indexes GLOBAL positions
    gx = pos[3 * (size_t)gi + 0];
    gy = pos[3 * (size_t)gi + 1];
    gz = pos[3 * (size_t)gi + 2];
    if (p + 4096 < P) __builtin_prefetch(&pidx[p + 4096], 0, 0);
  }
#if HAVE_ASYNC_LDS
  __builtin_amdgcn_s_wait_asynccnt(0);
  if (act) { rx = s_ref[3 * t + 0]; ry = s_ref[3 * t + 1]; rz = s_ref[3 * t + 2]; }
#else
  if (act) {
    rx = refp[3 * (size_t)p + 0];
    ry = refp[3 * (size_t)p + 1];
    rz = refp[3 * (size_t)p + 2];
  }
#endif
  if (act) {
    atomicAdd(&s[3 * c + 0], gx);
    atomicAdd(&s[3 * c + 1], gy);
    atomicAdd(&s[3 * c + 2], gz);
    atomicAdd(&s[3 * MAXC + 3 * c + 0], rx);
    atomicAdd(&s[3 * MAXC + 3 * c + 1], ry);
    atomicAdd(&s[3 * MAXC + 3 * c + 2], rz);
    atomicAdd(&s[6 * MAXC + c], 1.0f);
  }
  __syncthreads();
  float* o = partA + (size_t)blockIdx.x * NQ;
  for (int i = t; i < NQ; i += BLK) o[i] = s[i];
}

// Pass 2: reduce partials -> per-chain COMs -> best periodic translation.
__global__ void k_best_t(const float* __restrict__ partA,
                         const float* __restrict__ box,
                         const int*   __restrict__ nchp,
                         float* __restrict__ hdr, int nb) {
  __shared__ float s[NQ];
  const int t = threadIdx.x;
  for (int j = t; j < NQ; j += BLK) {
    float acc = 0.0f;
    for (int b = 0; b < nb; ++b) acc += partA[(size_t)b * NQ + j];
    s[j] = acc;
  }
  __syncthreads();
  const int nch = *nchp;
  if (t < nch) {
    const float invc = 1.0f / s[6 * MAXC + t];
    const float pcx = s[3 * t + 0] * invc, pcy = s[3 * t + 1] * invc, pcz = s[3 * t + 2] * invc;
    const float rcx = s[3 * MAXC + 3 * t + 0] * invc,
                rcy = s[3 * MAXC + 3 * t + 1] * invc,
                rcz = s[3 * MAXC + 3 * t + 2] * invc;
    float dx = rcx - pcx, dy = rcy - pcy, dz = rcz - pcz;  // ref_com - poc_com
    const float b0x = box[0], b0y = box[1], b0z = box[2];
    const float b1x = box[3], b1y = box[4], b1z = box[5];
    const float b2x = box[6], b2y = box[7], b2z = box[8];
    const float i0 = 1.0f / box[0], i1 = 1.0f / box[4], i2 = 1.0f / box[8];
    const float s3 = rintf(dz * i2);           // jnp.round = half-to-even = rintf(RTE)
    dx -= s3 * b2x; dy -= s3 * b2y; dz -= s3 * b2z;
    const float s2 = rintf(dy * i1);
    dx -= s2 * b1x; dy -= s2 * b1y; dz -= s2 * b1z;
    const float s1 = rintf(dx * i0);
    hdr[12 + 3 * t + 0] = s1 * b0x + s2 * b1x + s3 * b2x;
    hdr[12 + 3 * t + 1] = s1 * b0y + s2 * b1y + s3 * b2y;
    hdr[12 + 3 * t + 2] = s1 * b0z + s2 * b1z + s3 * b2z;
  }
}

// Pass 3: rec_pos = positions[rec_indices] + best_t[chain]; store + origin partials.
__global__ void k_rec_pos(const float* __restrict__ pos,
                          const int*   __restrict__ ridx,
                          const int*   __restrict__ cid,
                          const float* __restrict__ hdr,
                          float* __restrict__ recpos,
                          float* __restrict__ partB, int R) {
  __shared__ float sbt[3 * MAXC];
  __shared__ float sm[8];
  const int t = threadIdx.x;
  if (t < 3 * MAXC) sbt[t] = hdr[12 + t];
  __syncthreads();
  const int i = blockIdx.x * BLK + t;
  float x = 0.f, y = 0.f, z = 0.f;
  if (i < R) {
    if (i + 4096 < R) {
      __builtin_prefetch(&ridx[i + 4096], 0, 0);
      __builtin_prefetch(&cid[i + 4096], 0, 0);
    }
    const int g = ridx[i];
    const int c = cid[i];
    x = pos[3 * (size_t)g + 0] + sbt[3 * c + 0];
    y = pos[3 * (size_t)g + 1] + sbt[3 * c + 1];
    z = pos[3 * (size_t)g + 2] + sbt[3 * c + 2];
    recpos[3 * (size_t)i + 0] = x;
    recpos[3 * (size_t)i + 1] = y;
    recpos[3 * (size_t)i + 2] = z;
  }
  x = block_sum(x, sm); y = block_sum(y, sm); z = block_sum(z, sm);
  if (t == 0) {
    float* o = partB + 3 * (size_t)blockIdx.x;
    o[0] = x; o[1] = y; o[2] = z;
  }
}

__global__ void k_reduce3(const float* __restrict__ part, float* __restrict__ hdr,
                          int nb, int R) {
  __shared__ float sm[8];
  const int t = threadIdx.x;
  float sx = 0.f, sy = 0.f, sz = 0.f;
  for (int b = t; b < nb; b += BLK) {
    sx += part[3 * (size_t)b + 0];
    sy += part[3 * (size_t)b + 1];
    sz += part[3 * (size_t)b + 2];
  }
  sx = block_sum(sx, sm); sy = block_sum(sy, sm); sz = block_sum(sz, sm);
  if (t == 0) {
    const float inv = 1.0f / (float)R;
    hdr[0] = sx * inv; hdr[1] = sy * inv; hdr[2] = sz * inv;  // origin
  }
}

// Pass 5: pocket forces + torques (only pocket rows of F are nonzero).
__global__ void k_pocket_force(const float* __restrict__ recpos,
                               const float* __restrict__ refp,
                               const int*   __restrict__ pidx,
                               const float* __restrict__ kptr,
                               const float* __restrict__ hdr,
                               float* __restrict__ partC, int P) {
  __shared__ float sm[8];
  const int t = threadIdx.x;
  const int p = blockIdx.x * BLK + t;
  float Fx = 0.f, Fy = 0.f, Fz = 0.f, Tx = 0.f, Ty = 0.f, Tz = 0.f;
  if (p < P) {
    if (p + 4096 < P) __builtin_prefetch(&refp[3 * (size_t)(p + 4096)], 0, 0);
    const float k2 = -2.0f * kptr[0];
    const int ri = pidx[p];
    const float px = recpos[3 * (size_t)ri + 0];
    const float py = recpos[3 * (size_t)ri + 1];
    const float pz = recpos[3 * (size_t)ri + 2];
    const float fx = k2 * (px - refp[3 * (size_t)p + 0]);
    const float fy = k2 * (py - refp[3 * (size_t)p + 1]);
    const float fz = k2 * (pz - refp[3 * (size_t)p + 2]);
    const float cx = px - hdr[0], cy = py - hdr[1], cz = pz - hdr[2];
    Tx = cy * fz - cz * fy;     // cross(rec_centered, F)
    Ty = cz * fx - cx * fz;
    Tz = cx * fy - cy * fx;
    Fx = fx; Fy = fy; Fz = fz;
  }
  Fx = block_sum(Fx, sm); Fy = block_sum(Fy, sm); Fz = block_sum(Fz, sm);
  Tx = block_sum(Tx, sm); Ty = block_sum(Ty, sm); Tz = block_sum(Tz, sm);
  if (t == 0) {
    float* o = partC + 6 * (size_t)blockIdx.x;
    o[0] = Fx; o[1] = Fy; o[2] = Fz; o[3] = Tx; o[4] = Ty; o[5] = Tz;
  }
}

__global__ void k_reduce6(const float* __restrict__ part, float* __restrict__ hdr,
                          int nb, int R) {
  __shared__ float sm[8];
  const int t = threadIdx.x;
  float a0 = 0.f, a1 = 0.f, a2 = 0.f, a3 = 0.f, a4 = 0.f, a5 = 0.f;
  for (int b = t; b < nb; b += BLK) {
    const float* q = part + 6 * (size_t)b;
    a0 += q[0]; a1 += q[1]; a2 += q[2]; a3 += q[3]; a4 += q[4]; a5 += q[5];
  }
  a0 = block_sum(a0, sm); a1 = block_sum(a1, sm); a2 = block_sum(a2, sm);
  a3 = block_sum(a3, sm); a4 = block_sum(a4, sm); a5 = block_sum(a5, sm);
  if (t == 0) {
    const float inv = 1.0f / (float)R;          // means over ALL rec rows
    hdr[3] = a0 * inv; hdr[4] = a1 * inv; hdr[5] = a2 * inv;   // F_mean
    hdr[6] = a3 * inv; hdr[7] = a4 * inv; hdr[8] = a5 * inv;   // torque_mean
  }
}

// Pass 7: scatter F_mean + cross(torque_mean, rc)/r^2 into the global output.
__global__ void k_output(const float* __restrict__ recpos,
                         const int*   __restrict__ ridx,
                         const float* __restrict__ hdr,
                         float* __restrict__ out, int R) {
  const int i = blockIdx.x * BLK + threadIdx.x;
  if (i >= R) return;
  if (i + 4096 < R) {
    __builtin_prefetch(&recpos[3 * (size_t)(i + 4096)], 0, 0);
    __builtin_prefetch(&ridx[i + 4096], 0, 0);
  }
  const float cx = recpos[3 * (size_t)i + 0] - hdr[0];
  const float cy = recpos[3 * (size_t)i + 1] - hdr[1];
  const float cz = recpos[3 * (size_t)i + 2] - hdr[2];
  const float inv_rsq = 1.0f / (cx * cx + cy * cy + cz * cz);
  const float Tx = hdr[6], Ty = hdr[7], Tz = hdr[8];
  const float rx = Ty * cz - Tz * cy;
  const float ry = Tz * cx - Tx * cz;
  const float rz = Tx * cy - Ty * cx;
  const int g = ridx[i];
  out[3 * (size_t)g + 0] = hdr[3] + rx * inv_rsq;
  out[3 * (size_t)g + 1] = hdr[4] + ry * inv_rsq;
  out[3 * (size_t)g + 2] = hdr[5] + rz * inv_rsq;
}

extern "C" void kernel_launch(void* const* d_in, const int* in_sizes, int n_in,
                              void* d_out, int out_size, void* d_ws, size_t ws_size,
                              hipStream_t stream) {
  (void)n_in; (void)ws_size;
  const float* positions = (const float*)d_in[0];
  const float* box       = (const float*)d_in[1];
  const float* ref_poc   = (const float*)d_in[2];
  const float* kptr      = (const float*)d_in[3];
  const int*   rec_idx   = (const int*)d_in[4];
  const int*   poc_idx   = (const int*)d_in[5];
  const int*   chain_ids = (const int*)d_in[6];
  const int*   poc_cids  = (const int*)d_in[7];
  const int*   nch_ptr   = (const int*)d_in[8];

  const int R = in_sizes[4];   // rec_indices count (2,000,000)
  const int P = in_sizes[5];   // poc_indices count (500,000)

  const int nbP = (P + BLK - 1) / BLK;
  const int nbR = (R + BLK - 1) / BLK;

  float* w      = (float*)d_ws;
  float* hdr    = w;                                // 64 floats
  float* partA  = w + 64;                           // nbP * NQ
  float* partB  = partA + (size_t)nbP * NQ;         // nbR * 3
  float* partC  = partB + (size_t)nbR * 3;          // nbP * 6
  float* recpos = partC + (size_t)nbP * 6;          // R * 3  (~24 MB, L2-resident)

  float* Fout = (float*)d_out;  // [0] = energy (0.0), [1..] = F_final flat

  const int n4 = out_size >> 2;                     // b128 zero fill
  if (n4 > 0) k_zero4<<<(n4 + BLK - 1) / BLK, BLK, 0, stream>>>((float4*)Fout, n4);
  if (out_size - (n4 << 2) > 0)
    k_zero_tail<<<1, BLK, 0, stream>>>(Fout, n4 << 2, out_size);

  k_pocket_com<<<nbP, BLK, 0, stream>>>(positions, ref_poc, poc_idx, poc_cids, partA, P);
  k_best_t<<<1, BLK, 0, stream>>>(partA, box, nch_ptr, hdr, nbP);
  k_rec_pos<<<nbR, BLK, 0, stream>>>(positions, rec_idx, chain_ids, hdr, recpos, partB, R);
  k_reduce3<<<1, BLK, 0, stream>>>(partB, hdr, nbR, R);
  k_pocket_force<<<nbP, BLK, 0, stream>>>(recpos, ref_poc, poc_idx, kptr, hdr, partC, P);
  k_reduce6<<<1, BLK, 0, stream>>>(partC, hdr, nbP, R);
  k_output<<<nbR, BLK, 0, stream>>>(recpos, rec_idx, hdr, Fout + 1, R);
}